// QuantumNATEnhanced_65481071400022
// MI455X (gfx1250) — compile-verified
//
#include <hip/hip_runtime.h>
#include <math.h>

typedef __attribute__((ext_vector_type(2))) float v2f;
typedef __attribute__((ext_vector_type(8))) float v8f;

#define CONV_BLOCK 256

// ---------------------------------------------------------------------------
// Kernel 1: per-image conv1+relu+pool2 -> conv2+relu+pool2 -> flat(196)
// One 256-thread block per image. relu(pool(y)) == pool-as-max(relu(y)),
// and relu(max(a..)) == max(relu(a)..), so pooling is fused into the conv.
// ---------------------------------------------------------------------------
__global__ void conv_fused_kernel(const float* __restrict__ x,
                                  const float* __restrict__ w1, const float* __restrict__ b1,
                                  const float* __restrict__ w2, const float* __restrict__ b2,
                                  float* __restrict__ flat) {
    __shared__ float img[28 * 28];       // input image
    __shared__ float h1[4 * 14 * 14];    // stage-1 pooled output

    const float* xin = x + (size_t)blockIdx.x * 784;
    for (int t = threadIdx.x; t < 784; t += CONV_BLOCK) img[t] = xin[t];
    __syncthreads();

    // Stage 1: conv1 (1->4 ch, 3x3 SAME on 28x28) + relu + 2x2 maxpool -> (4,14,14)
    for (int t = threadIdx.x; t < 4 * 14 * 14; t += CONV_BLOCK) {
        const int c = t / 196;
        const int r = t % 196;
        const int oh = r / 14, ow = r % 14;
        float mx = -1e30f;
        #pragma unroll
        for (int dy = 0; dy < 2; ++dy) {
            #pragma unroll
            for (int dx = 0; dx < 2; ++dx) {
                const int y = oh * 2 + dy, xx = ow * 2 + dx;
                float acc = b1[c];
                #pragma unroll
                for (int ky = 0; ky < 3; ++ky) {
                    const int iy = y + ky - 1;
                    if (iy < 0 || iy >= 28) continue;
                    #pragma unroll
                    for (int kx = 0; kx < 3; ++kx) {
                        const int ix = xx + kx - 1;
                        if (ix < 0 || ix >= 28) continue;
                        acc += img[iy * 28 + ix] * w1[c * 9 + ky * 3 + kx];
                    }
                }
                mx = fmaxf(mx, acc);
            }
        }
        h1[t] = fmaxf(mx, 0.0f);
    }
    __syncthreads();

    // Stage 2: conv2 (4->4 ch, 3x3 SAME on 14x14) + relu + 2x2 maxpool -> (4,7,7)
    float* frow = flat + (size_t)blockIdx.x * 196;
    for (int t = threadIdx.x; t < 4 * 7 * 7; t += CONV_BLOCK) {
        const int c = t / 49;
        const int r = t % 49;
        const int oh = r / 7, ow = r % 7;
        float mx = -1e30f;
        #pragma unroll
        for (int dy = 0; dy < 2; ++dy) {
            #pragma unroll
            for (int dx = 0; dx < 2; ++dx) {
                const int y = oh * 2 + dy, xx = ow * 2 + dx;
                float acc = b2[c];
                #pragma unroll
                for (int ci = 0; ci < 4; ++ci) {
                    #pragma unroll
                    for (int ky = 0; ky < 3; ++ky) {
                        const int iy = y + ky - 1;
                        if (iy < 0 || iy >= 14) continue;
                        #pragma unroll
                        for (int kx = 0; kx < 3; ++kx) {
                            const int ix = xx + kx - 1;
                            if (ix < 0 || ix >= 14) continue;
                            acc += h1[ci * 196 + iy * 14 + ix] * w2[(c * 4 + ci) * 9 + ky * 3 + kx];
                        }
                    }
                }
                mx = fmaxf(mx, acc);
            }
        }
        frow[t] = fmaxf(mx, 0.0f);
    }
}

// ---------------------------------------------------------------------------
// Kernel 2: params(B,16) = flat(B,196) @ pl_w.T(196,16) + pl_b
// One wave per 16-row batch tile; K=196 = 49 chained V_WMMA_F32_16X16X4_F32.
// fp32 WMMA -> exact fp32 math, no precision loss vs reference.
// Fragment layouts (ISA 7.12.2, 32-bit):
//   A(m,k): lane = m + 16*(k>>1), vgpr = k&1
//   B(k,n): lane = n + 16*(k>>1), vgpr = k&1
//   D:      vgpr r holds M = r + 8*(lane>>4), N = lane&15
// ---------------------------------------------------------------------------
__global__ void gemm_wmma_kernel(const float* __restrict__ flat,
                                 const float* __restrict__ pl_w,
                                 const float* __restrict__ pl_b,
                                 float* __restrict__ params) {
    const int wave = threadIdx.x >> 5;
    const int lane = threadIdx.x & 31;
    const int tile = blockIdx.x * (blockDim.x >> 5) + wave;   // 16 batch rows per tile
    const int half = lane >> 4;   // selects k-pair {0,1} vs {2,3} within a K=4 block
    const int idx  = lane & 15;   // A: row m   /  B: column n

    const float* arow = flat + ((size_t)tile * 16 + idx) * 196 + 2 * half;
    const float* brow = pl_w + (size_t)idx * 196 + 2 * half;   // B(k,n) = pl_w[n][k]

    v8f acc = {};
    #pragma unroll 7
    for (int kk = 0; kk < 49; ++kk) {
        v2f a, b;
        a.x = arow[kk * 4 + 0];
        a.y = arow[kk * 4 + 1];
        b.x = brow[kk * 4 + 0];
        b.y = brow[kk * 4 + 1];
        acc = __builtin_amdgcn_wmma_f32_16x16x4_f32(
            /*neg_a=*/false, a, /*neg_b=*/false, b,
            /*c_mod=*/(short)0, acc, /*reuse_a=*/false, /*reuse_b=*/false);
    }

    const float bias = pl_b[idx];
    float* out = params + ((size_t)tile * 16 + 8 * half) * 16 + idx;
    #pragma unroll
    for (int r = 0; r < 8; ++r) out[r * 16] = acc[r] + bias;
}

// ---------------------------------------------------------------------------
// Kernel 3: 4-qubit statevector circuit + readout, one thread per sample.
// Also produces deterministic per-block (sum, sumsq) partials for BN.
// Index convention: j = w0*8 + w1*4 + w2*2 + w3  (wire i <-> stride 8>>i)
// ---------------------------------------------------------------------------
__global__ void circuit_kernel(const float* __restrict__ params,
                               const float* __restrict__ ro_w,
                               const float* __restrict__ ro_b,
                               float* __restrict__ out_raw,
                               float* __restrict__ partials) {
    const int i = blockIdx.x * blockDim.x + threadIdx.x;

    float pr[16], pi[16];
    #pragma unroll
    for (int j = 0; j < 16; ++j) { pr[j] = 0.0f; pi[j] = 0.0f; }
    pr[0] = 1.0f;

    const float* P = params + (size_t)i * 16;

    #pragma unroll
    for (int w = 0; w < 4; ++w) {
        const int s = 8 >> w;                 // stride of wire w
        // ---- RY(theta0): real rotation [[c,-s],[s,c]] ----
        {
            const float th = 0.5f * P[4 * w + 0];
            const float c = __cosf(th), sn = __sinf(th);
            #pragma unroll
            for (int a = 0; a < 16; ++a) {
                if (a & s) continue;
                const int b = a + s;
                float r0 = pr[a], i0 = pi[a], r1 = pr[b], i1 = pi[b];
                pr[a] = c * r0 - sn * r1;  pi[a] = c * i0 - sn * i1;
                pr[b] = sn * r0 + c * r1;  pi[b] = sn * i0 + c * i1;
            }
        }
        // ---- RZ(theta1): diag(e^{-i th}, e^{+i th}) ----
        {
            const float th = 0.5f * P[4 * w + 1];
            const float c = __cosf(th), sn = __sinf(th);
            #pragma unroll
            for (int a = 0; a < 16; ++a) {
                if (a & s) continue;
                const int b = a + s;
                float r0 = pr[a], i0 = pi[a], r1 = pr[b], i1 = pi[b];
                pr[a] = c * r0 + sn * i0;  pi[a] = c * i0 - sn * r0;
                pr[b] = c * r1 - sn * i1;  pi[b] = c * i1 + sn * r1;
            }
        }
        // ---- RX(theta2): [[c, -i s],[-i s, c]] ----
        {
            const float th = 0.5f * P[4 * w + 2];
            const float c = __cosf(th), sn = __sinf(th);
            #pragma unroll
            for (int a = 0; a < 16; ++a) {
                if (a & s) continue;
                const int b = a + s;
                float r0 = pr[a], i0 = pi[a], r1 = pr[b], i1 = pi[b];
                pr[a] = c * r0 + sn * i1;  pi[a] = c * i0 - sn * r1;
                pr[b] = sn * i0 + c * r1;  pi[b] = -sn * r0 + c * i1;
            }
        }
        // ---- CNOT(control=w, target=(w+1)%4): flip target bit where control=1 ----
        {
            const int sc = 8 >> w;
            const int st = 8 >> ((w + 1) & 3);
            #pragma unroll
            for (int a = 0; a < 16; ++a) {
                if (!(a & sc) || (a & st)) continue;
                const int b = a + st;
                float tr = pr[a]; pr[a] = pr[b]; pr[b] = tr;
                float ti = pi[a]; pi[a] = pi[b]; pi[b] = ti;
            }
        }
    }

    // PauliZ expvals
    float z[4] = {0.f, 0.f, 0.f, 0.f};
    #pragma unroll
    for (int j = 0; j < 16; ++j) {
        const float p = pr[j] * pr[j] + pi[j] * pi[j];
        #pragma unroll
        for (int w = 0; w < 4; ++w)
            z[w] += (j & (8 >> w)) ? -p : p;
    }

    // Readout: out[k] = z . ro_w[k][:] + ro_b[k]
    float o[4];
    #pragma unroll
    for (int k = 0; k < 4; ++k) {
        float acc = ro_b[k];
        #pragma unroll
        for (int q = 0; q < 4; ++q) acc += z[q] * ro_w[k * 4 + q];
        o[k] = acc;
        out_raw[(size_t)i * 4 + k] = acc;
    }

    // Deterministic block reduction of (sum, sumsq) per output column.
    __shared__ float sdata[256];
    #pragma unroll
    for (int q = 0; q < 8; ++q) {
        const float v = (q < 4) ? o[q] : o[q - 4] * o[q - 4];
        sdata[threadIdx.x] = v;
        __syncthreads();
        for (int sft = 128; sft > 0; sft >>= 1) {
            if ((int)threadIdx.x < sft) sdata[threadIdx.x] += sdata[threadIdx.x + sft];
            __syncthreads();
        }
        if (threadIdx.x == 0) partials[blockIdx.x * 8 + q] = sdata[0];
        __syncthreads();
    }
}

// ---------------------------------------------------------------------------
// Kernel 4: fold block partials -> BN scale/shift (biased variance)
// ---------------------------------------------------------------------------
__global__ void stats_kernel(const float* __restrict__ partials, int nblocks, float invB,
                             const float* __restrict__ bn_w, const float* __restrict__ bn_b,
                             float* __restrict__ stats) {
    const int k = threadIdx.x;
    if (k < 4) {
        float s = 0.0f, sq = 0.0f;
        for (int b = 0; b < nblocks; ++b) {
            s  += partials[b * 8 + k];
            sq += partials[b * 8 + 4 + k];
        }
        const float mean = s * invB;
        const float var  = sq * invB - mean * mean;
        const float scale = bn_w[k] * rsqrtf(var + 1e-5f);
        stats[k]     = scale;
        stats[4 + k] = bn_b[k] - mean * scale;
    }
}

// ---------------------------------------------------------------------------
// Kernel 5: apply batchnorm
// ---------------------------------------------------------------------------
__global__ void bn_apply_kernel(const float* __restrict__ out_raw,
                                const float* __restrict__ stats,
                                float* __restrict__ out, int n) {
    const int i = blockIdx.x * blockDim.x + threadIdx.x;
    if (i < n) {
        const int k = i & 3;
        out[i] = out_raw[i] * stats[k] + stats[4 + k];
    }
}

// ---------------------------------------------------------------------------
extern "C" void kernel_launch(void* const* d_in, const int* in_sizes, int n_in,
                              void* d_out, int out_size, void* d_ws, size_t ws_size,
                              hipStream_t stream) {
    const float* x       = (const float*)d_in[0];
    const float* conv1_w = (const float*)d_in[1];
    const float* conv1_b = (const float*)d_in[2];
    const float* conv2_w = (const float*)d_in[3];
    const float* conv2_b = (const float*)d_in[4];
    const float* pl_w    = (const float*)d_in[5];
    const float* pl_b    = (const float*)d_in[6];
    const float* ro_w    = (const float*)d_in[7];
    const float* ro_b    = (const float*)d_in[8];
    const float* bn_w    = (const float*)d_in[9];
    const float* bn_b    = (const float*)d_in[10];

    const int B = in_sizes[0] / (28 * 28);   // 16384

    // Workspace layout (floats)
    float* ws       = (float*)d_ws;
    float* flat     = ws;                          // B*196
    float* params   = flat + (size_t)B * 196;      // B*16
    float* out_raw  = params + (size_t)B * 16;     // B*4
    const int nred  = B / 256;                     // circuit blocks (64)
    float* partials = out_raw + (size_t)B * 4;     // nred*8
    float* stats    = partials + (size_t)nred * 8; // 8

    // 1. conv stack -> flat
    conv_fused_kernel<<<B, CONV_BLOCK, 0, stream>>>(x, conv1_w, conv1_b, conv2_w, conv2_b, flat);

    // 2. flat @ pl_w.T + pl_b via f32 WMMA (one wave per 16-row tile, 8 waves/block)
    const int tiles = B / 16;                // 1024
    gemm_wmma_kernel<<<tiles / 8, 256, 0, stream>>>(flat, pl_w, pl_b, params);

    // 3. quantum circuit + readout + BN partial sums
    circuit_kernel<<<nred, 256, 0, stream>>>(params, ro_w, ro_b, out_raw, partials);

    // 4. BN statistics
    stats_kernel<<<1, 32, 0, stream>>>(partials, nred, 1.0f / (float)B, bn_w, bn_b, stats);

    // 5. normalize
    const int n = B * 4;
    bn_apply_kernel<<<(n + 255) / 256, 256, 0, stream>>>(out_raw, stats, (float*)d_out, n);
}